// SAGE_69724499083377
// MI455X (gfx1250) — compile-verified
//
#include <hip/hip_runtime.h>
#include <hip/hip_bf16.h>

#define N_NODES 100000
#define N_EDGES 1600000
#define IN_CH   64
#define HID_CH  128
// Packed-weight LDS layout: float2 pair (W[2kk][c], W[2kk+1][c]) stored at
// float index kk*PKS + c*2 + j.  PKS=288 dwords: 288 % 64 == 32, so the two
// half-waves of a b64 fragment read disjoint 32-bank sets -> conflict-free.
#define PKS     288

typedef __attribute__((ext_vector_type(2))) float v2f;
typedef __attribute__((ext_vector_type(4))) float v4f;
typedef __attribute__((ext_vector_type(8))) float v8f;

// ---------------------------------------------------------------------------
// Kernel 1: edge scatter. 64 threads per edge -> coalesced 256B gather of the
// source row and coalesced f32 atomics into agg[dst]. Lane c==0 counts degree.
// Edge indices are streamed once -> non-temporal loads keep L2 for h/agg.
// ---------------------------------------------------------------------------
__global__ __launch_bounds__(256) void sage_scatter(
    const int* __restrict__ ei,      // [2, N_EDGES] (dst row 0, src row 1)
    const float* __restrict__ h,     // [N_NODES, 64]
    float* __restrict__ agg,         // [N_NODES, 64]
    float* __restrict__ deg)         // [N_NODES]
{
    unsigned tid = blockIdx.x * 256u + threadIdx.x;
    unsigned e = tid >> 6;           // edge id
    unsigned c = tid & 63u;          // channel
    if (e >= N_EDGES) return;
    int dst = __builtin_nontemporal_load(ei + e);
    int src = __builtin_nontemporal_load(ei + N_EDGES + e);
    float val = h[(size_t)src * IN_CH + c];
    atomicAdd(&agg[(size_t)dst * IN_CH + c], val);
    if (c == 0) atomicAdd(&deg[dst], 1.0f);
}

// ---------------------------------------------------------------------------
// Kernel 2: fused mean-normalize + dual GEMM + bias using f32 WMMA.
// Weights are staged once per block into LDS in K-paired layout so each WMMA
// B fragment is one conflict-free ds_load_b64 with an immediate offset.
// One wave32 owns a 16-row strip of nodes:
//   out[16x128] = (agg/deg)[16x64] @ Wl[64x128] + h[16x64] @ Wr[64x128] + b
// ---------------------------------------------------------------------------
__global__ __launch_bounds__(256) void sage_gemm(
    const float* __restrict__ agg,   // [N_NODES, 64]
    const float* __restrict__ deg,   // [N_NODES]
    const float* __restrict__ h,     // [N_NODES, 64]
    const float* __restrict__ wl,    // [64, 128]
    const float* __restrict__ wr,    // [64, 128]
    const float* __restrict__ bias,  // [128]
    float* __restrict__ out)         // [N_NODES, 128]
{
    __shared__ float wpk[2][32 * PKS];              // 72 KB of the 320 KB WGP LDS

    // Cooperative staging: coalesced b128 global loads, repacked so that
    // consecutive K-pairs of one column are adjacent (one b64 per fragment).
    {
        int tid = threadIdx.x;
#pragma unroll
        for (int it = 0; it < 8; ++it) {
            int d = it * 1024 + tid * 4;            // flat dword idx into [64,128]
            int r = d >> 7;                         // K row
            int c = d & 127;                        // column (4-aligned)
            v4f a = *(const v4f*)(wl + d);
            float* p = &wpk[0][(r >> 1) * PKS + (r & 1)];
            p[(c + 0) * 2] = a.x; p[(c + 1) * 2] = a.y;
            p[(c + 2) * 2] = a.z; p[(c + 3) * 2] = a.w;
            v4f bq = *(const v4f*)(wr + d);
            float* q = &wpk[1][(r >> 1) * PKS + (r & 1)];
            q[(c + 0) * 2] = bq.x; q[(c + 1) * 2] = bq.y;
            q[(c + 2) * 2] = bq.z; q[(c + 3) * 2] = bq.w;
        }
    }
    __syncthreads();

    const int NTILES_M = N_NODES / 16;              // 6250, exact
    int wave = blockIdx.x * 8 + (threadIdx.x >> 5); // uniform per wave
    if (wave >= NTILES_M) return;                   // uniform: EXEC stays all-1s below
    int lane = threadIdx.x & 31;
    int half = lane >> 4;                           // 0: K even pair, 1: K odd pair
    int lrow = lane & 15;                           // matrix row M within tile
    int row  = wave * 16 + lrow;

    // per-row mean scale (lane-uniform across the two half-wave copies of a row)
    float s = 1.0f / fmaxf(deg[row], 1.0f);

    // Preload A fragments for the whole K=64 strip (ISA 32-bit 16x4 A layout):
    // lanes 0-15 hold (K=4k, 4k+1), lanes 16-31 hold (K=4k+2, 4k+3) of row M=lrow.
    const float* ap = agg + (size_t)row * IN_CH + half * 2;
    const float* hp = h   + (size_t)row * IN_CH + half * 2;
    v2f a_mean[16], a_h[16];
#pragma unroll
    for (int k = 0; k < 16; ++k) {
        v2f am = *(const v2f*)(ap + 4 * k);
        a_mean[k] = am * s;                         // fold mean division into A
        a_h[k]    = *(const v2f*)(hp + 4 * k);
    }

#pragma unroll 1
    for (int n = 0; n < HID_CH / 16; ++n) {         // 8 column tiles
        int col = n * 16 + lrow;
        // B fragment for k-step k, matrix m: float2 at wpk[m][(2k+half)*PKS + col*2]
        const float* b0 = &wpk[0][half * PKS + col * 2];
        const float* b1 = &wpk[1][half * PKS + col * 2];
        v8f c = {0.f, 0.f, 0.f, 0.f, 0.f, 0.f, 0.f, 0.f};
#pragma unroll
        for (int k = 0; k < 16; ++k) {              // mean @ Wl
            v2f b = *(const v2f*)(b0 + k * (2 * PKS));
            c = __builtin_amdgcn_wmma_f32_16x16x4_f32(
                    false, a_mean[k], false, b, (short)0, c, false, false);
        }
#pragma unroll
        for (int k = 0; k < 16; ++k) {              // h @ Wr (same accumulator)
            v2f b = *(const v2f*)(b1 + k * (2 * PKS));
            c = __builtin_amdgcn_wmma_f32_16x16x4_f32(
                    false, a_h[k], false, b, (short)0, c, false, false);
        }
        float bv = bias[col];
        // C/D layout: VGPR v -> M = v (lanes 0-15) / v+8 (lanes 16-31), N = lane%16
#pragma unroll
        for (int v = 0; v < 8; ++v) {
            int m = wave * 16 + v + half * 8;
            __builtin_nontemporal_store(c[v] + bv, &out[(size_t)m * HID_CH + col]);
        }
    }
}

// ---------------------------------------------------------------------------
extern "C" void kernel_launch(void* const* d_in, const int* in_sizes, int n_in,
                              void* d_out, int out_size, void* d_ws, size_t ws_size,
                              hipStream_t stream) {
    // inputs: 0=x (unused), 1=edge_index, 2=emb_weight, 3=lin_l_w, 4=lin_l_b, 5=lin_r_w
    const int*   ei = (const int*)d_in[1];
    const float* h  = (const float*)d_in[2];
    const float* wl = (const float*)d_in[3];
    const float* bi = (const float*)d_in[4];
    const float* wr = (const float*)d_in[5];
    float* out = (float*)d_out;

    float* agg = (float*)d_ws;                       // [N_NODES, 64]
    float* deg = agg + (size_t)N_NODES * IN_CH;      // [N_NODES]

    // zero agg + deg (capture-legal async memset)
    hipMemsetAsync(d_ws, 0, (size_t)N_NODES * (IN_CH + 1) * sizeof(float), stream);

    // scatter: 64 threads per edge
    {
        unsigned total = (unsigned)N_EDGES * 64u;    // 102.4M threads
        unsigned blocks = (total + 255u) / 256u;
        sage_scatter<<<blocks, 256, 0, stream>>>(ei, h, agg, deg);
    }

    // fused mean + dual GEMM + bias: one wave per 16-row strip
    {
        int waves  = N_NODES / 16;                   // 6250
        int blocks = (waves + 7) / 8;                // 8 waves per block
        sage_gemm<<<blocks, 256, 0, stream>>>(agg, deg, h, wl, wr, bi, out);
    }
}